// GAT_67336497266745
// MI455X (gfx1250) — compile-verified
//
#include <hip/hip_runtime.h>
#include <hip/hip_bf16.h>
#include <math.h>

#define NFEAT  256
#define HEADS  8
#define NHID   8
#define HD     (HEADS * NHID)   // 64
#define NCLASS 16
#define LAMBDA1 0.5f
#define LAMBDA2 0.5f
#define SLOPE   0.2f

typedef float v2f __attribute__((ext_vector_type(2)));
typedef float v8f __attribute__((ext_vector_type(8)));
typedef unsigned int v4u __attribute__((ext_vector_type(4)));
typedef unsigned int v8u __attribute__((ext_vector_type(8)));

// D = A(16x4 f32) * B(4x16 f32) + C(16x16 f32), full fp32 path on CDNA5.
static __device__ __forceinline__ v8f wmma_f32_k4(v2f a, v2f b, v8f c) {
  return __builtin_amdgcn_wmma_f32_16x16x4_f32(
      /*neg_a=*/false, a, /*neg_b=*/false, b,
      /*c_mod=*/(short)0, c, /*reuse_a=*/false, /*reuse_b=*/false);
}

// Allocation-relative LDS byte offset of a __shared__ object (low 32 bits of
// the generic pointer are the LDS offset on gfx12xx).
static __device__ __forceinline__ unsigned lds_byte_off(const void* p) {
  return (unsigned)(unsigned long long)p;
}

// Issue one TDM transfer: nelem f32, contiguous, global -> LDS.
// D# per CDNA5 ISA ch.8: group0 = {count|flags, lds_addr, ga_lo, ga_hi|type},
// group1 = {mask/data_size, dims/strides}. 2-D descriptor => groups 2/3 NULL.
static __device__ __forceinline__ void tdm_load_to_lds(unsigned lds_off,
                                                       const void* gptr,
                                                       unsigned nelem) {
  unsigned long long ga = (unsigned long long)gptr;
  v4u g0;
  g0.x = 1u;                                         // count=1, is_restore=0
  g0.y = lds_off;                                    // lds_addr (bytes)
  g0.z = (unsigned)(ga & 0xFFFFFFFFu);               // global_addr[31:0]
  g0.w = (unsigned)((ga >> 32) & 0x01FFFFFFu)        // global_addr[56:32]
       | (2u << 30);                                 // type=2 ("image")
  v8u g1;
  g1.s0 = (2u << 16);                                // data_size=2 -> 4 bytes
  g1.s1 = (nelem & 0xFFFFu) << 16;                   // tensor_dim0[15:0]
  g1.s2 = ((nelem >> 16) & 0xFFFFu) | (1u << 16);    // tensor_dim0[31:16], tensor_dim1=1
  g1.s3 = (nelem & 0xFFFFu) << 16;                   // tile_dim0 (<=65535 elems)
  g1.s4 = 1u;                                        // tile_dim1=1, tile_dim2=0
  g1.s5 = nelem;                                     // tensor_dim0_stride[31:0]
  g1.s6 = 0u;                                        // stride hi / dim1_stride lo
  g1.s7 = 0u;                                        // dim1_stride hi
  asm volatile("tensor_load_to_lds %0, %1" :: "s"(g0), "s"(g1) : "memory");
}

// -------------------- GEMM1: h1[N,64] = x[N,256] @ W1[256,64] --------------------
// 8 waves / block: 2 M-tiles x 4 N-tiles. W1 (64KB) staged in LDS by the TDM.
__global__ __launch_bounds__(256)
void gemm1_wmma(const float* __restrict__ x, const float* __restrict__ W,
                float* __restrict__ h1, int N, int mtiles) {
  __shared__ float sW[NFEAT * HD];  // 64 KB
  if (threadIdx.x < 32) {           // wave 0 drives the DMA
    tdm_load_to_lds(lds_byte_off(sW), W, NFEAT * HD);
    __builtin_amdgcn_s_wait_tensorcnt(0);
  }
  __syncthreads();

  const int wave  = threadIdx.x >> 5;
  const int lane  = threadIdx.x & 31;
  const int mtile = blockIdx.x * 2 + (wave >> 2);
  if (mtile >= mtiles) return;              // wave-uniform; EXEC stays full for WMMA
  const int ntile = wave & 3;
  const int m     = lane & 15;
  const int koff  = (lane >> 4) << 1;
  const int half8 = (lane >> 4) << 3;
  const int nb    = ntile * 16 + m;         // output column for this lane (B/C role)

  const float* __restrict__ xrow = x + (size_t)(mtile * 16 + m) * NFEAT;

  v8f c0 = {}, c1 = {};                     // two chains -> 2x WMMA ILP
  #pragma unroll 4
  for (int k = 0; k < NFEAT; k += 8) {
    v2f a, b;
    a.x = xrow[k + koff];
    a.y = xrow[k + koff + 1];
    b.x = sW[(k + koff)     * HD + nb];
    b.y = sW[(k + koff + 1) * HD + nb];
    c0 = wmma_f32_k4(a, b, c0);
    a.x = xrow[k + 4 + koff];
    a.y = xrow[k + 4 + koff + 1];
    b.x = sW[(k + 4 + koff)     * HD + nb];
    b.y = sW[(k + 4 + koff + 1) * HD + nb];
    c1 = wmma_f32_k4(a, b, c1);
  }
  v8f c = c0 + c1;
  float* __restrict__ dst = h1 + (size_t)(mtile * 16 + half8) * HD + nb;
  #pragma unroll
  for (int r = 0; r < 8; ++r) dst[(size_t)r * HD] = c[r];
}

// ---- GEMM2: h2[N,16] = h[N,64] @ W2[64,16]; side2 = h @ fc2_W[64,16] (fused) ----
__global__ __launch_bounds__(256)
void gemm2_wmma(const float* __restrict__ h, const float* __restrict__ W2,
                const float* __restrict__ fcW, float* __restrict__ h2,
                float* __restrict__ side2, int N, int mtiles) {
  __shared__ float sB[2 * HD * NCLASS];     // 8 KB: [W2 | fc2_W]
  if (threadIdx.x < 32) {
    tdm_load_to_lds(lds_byte_off(sB), W2, HD * NCLASS);
    tdm_load_to_lds(lds_byte_off(sB) + HD * NCLASS * sizeof(float), fcW, HD * NCLASS);
    __builtin_amdgcn_s_wait_tensorcnt(0);
  }
  __syncthreads();

  const int wave  = threadIdx.x >> 5;
  const int lane  = threadIdx.x & 31;
  const int mtile = blockIdx.x * 8 + wave;
  if (mtile >= mtiles) return;
  const int m     = lane & 15;
  const int koff  = (lane >> 4) << 1;
  const int half8 = (lane >> 4) << 3;

  const float* __restrict__ hrow = h + (size_t)(mtile * 16 + m) * HD;
  const float* __restrict__ sW2 = sB;
  const float* __restrict__ sFc = sB + HD * NCLASS;

  v8f c0 = {}, c1 = {};
  #pragma unroll
  for (int k = 0; k < HD; k += 4) {
    v2f a;
    a.x = hrow[k + koff];
    a.y = hrow[k + koff + 1];
    v2f b0, b1v;
    b0.x  = sW2[(k + koff)     * NCLASS + m];
    b0.y  = sW2[(k + koff + 1) * NCLASS + m];
    b1v.x = sFc[(k + koff)     * NCLASS + m];
    b1v.y = sFc[(k + koff + 1) * NCLASS + m];
    c0 = wmma_f32_k4(a, b0,  c0);
    c1 = wmma_f32_k4(a, b1v, c1);
  }
  size_t base = (size_t)(mtile * 16 + half8) * NCLASS + m;
  #pragma unroll
  for (int r = 0; r < 8; ++r) {
    h2   [base + (size_t)r * NCLASS] = c0[r];
    side2[base + (size_t)r * NCLASS] = c1[r];
  }
}

// ------------- attention coefficients + fc1 side branch (layer 1) -------------
__global__ void attn1_kernel(const float* __restrict__ x, const float* __restrict__ h1,
                             const float* __restrict__ fc1W, const float* __restrict__ fc1b,
                             const float* __restrict__ a1s, const float* __restrict__ a1d,
                             float* __restrict__ as1, float* __restrict__ ad1,
                             float* __restrict__ side1, int N) {
  int n = blockIdx.x * blockDim.x + threadIdx.x;
  if (n >= N) return;
  float acc[NHID];
  #pragma unroll
  for (int c = 0; c < NHID; ++c) acc[c] = fc1b[c];
  const float* __restrict__ xr = x + (size_t)n * NFEAT;
  for (int k = 0; k < NFEAT; ++k) {
    float xv = xr[k];
    #pragma unroll
    for (int c = 0; c < NHID; ++c) acc[c] = fmaf(xv, fc1W[k * NHID + c], acc[c]);
  }
  #pragma unroll
  for (int c = 0; c < NHID; ++c) side1[(size_t)n * NHID + c] = acc[c];

  const float* __restrict__ hr = h1 + (size_t)n * HD;
  #pragma unroll
  for (int hh = 0; hh < HEADS; ++hh) {
    float ss = 0.f, dd = 0.f;
    #pragma unroll
    for (int c = 0; c < NHID; ++c) {
      float v = hr[hh * NHID + c];
      ss = fmaf(v, a1s[hh * NHID + c], ss);
      dd = fmaf(v, a1d[hh * NHID + c], dd);
    }
    as1[(size_t)n * HEADS + hh] = ss;
    ad1[(size_t)n * HEADS + hh] = dd;
  }
}

__global__ void attn2_kernel(const float* __restrict__ h2, const float* __restrict__ a2s,
                             const float* __restrict__ a2d, float* __restrict__ as2,
                             float* __restrict__ ad2, int N) {
  int n = blockIdx.x * blockDim.x + threadIdx.x;
  if (n >= N) return;
  const float* __restrict__ hr = h2 + (size_t)n * NCLASS;
  float ss = 0.f, dd = 0.f;
  #pragma unroll
  for (int c = 0; c < NCLASS; ++c) {
    ss = fmaf(hr[c], a2s[c], ss);
    dd = fmaf(hr[c], a2d[c], dd);
  }
  as2[n] = ss; ad2[n] = dd;
}

static __device__ __forceinline__ void edge_nodes(const int* __restrict__ ei, int E,
                                                  int idx, int& s, int& d) {
  if (idx < E) { s = ei[idx]; d = ei[E + idx]; }
  else { s = idx - E; d = s; }               // appended self loops
}

// ---- edge passes, layer 1 (softmax is shift-invariant: skip segment-max) ----
__global__ void edge1_pass1(const int* __restrict__ ei, int E, int Etot,
                            const float* __restrict__ as1, const float* __restrict__ ad1,
                            float* __restrict__ denom1) {
  int idx = blockIdx.x * blockDim.x + threadIdx.x;
  if (idx >= Etot) return;
  int s, d; edge_nodes(ei, E, idx, s, d);
  #pragma unroll
  for (int hh = 0; hh < HEADS; ++hh) {
    float e = as1[(size_t)s * HEADS + hh] + ad1[(size_t)d * HEADS + hh];
    e = e > 0.f ? e : SLOPE * e;
    atomicAdd(&denom1[(size_t)d * HEADS + hh], expf(e));
  }
}

__global__ void edge1_pass2(const int* __restrict__ ei, int E, int Etot,
                            const float* __restrict__ as1, const float* __restrict__ ad1,
                            const float* __restrict__ denom1, const float* __restrict__ h1,
                            float* __restrict__ out1) {
  int t = blockIdx.x * blockDim.x + threadIdx.x;   // (edge, head)
  int idx = t >> 3, hh = t & 7;
  if (idx >= Etot) return;
  int s, d; edge_nodes(ei, E, idx, s, d);
  float e = as1[(size_t)s * HEADS + hh] + ad1[(size_t)d * HEADS + hh];
  e = e > 0.f ? e : SLOPE * e;
  float alpha = expf(e) / denom1[(size_t)d * HEADS + hh];
  const float* __restrict__ hs = h1 + (size_t)s * HD + hh * NHID;
  float* __restrict__ od = out1 + (size_t)d * HD + hh * NHID;
  #pragma unroll
  for (int c = 0; c < NHID; ++c) atomicAdd(&od[c], hs[c] * alpha);
}

__global__ void elu_kernel(const float* __restrict__ out1, const float* __restrict__ side1,
                           const float* __restrict__ b1, float* __restrict__ hbuf,
                           long long total) {
  long long t = blockIdx.x * (long long)blockDim.x + threadIdx.x;
  if (t >= total) return;
  int col = (int)(t & (HD - 1));
  long long n = t >> 6;
  float v = out1[t] + b1[col] - LAMBDA1 * side1[n * NHID + (col & (NHID - 1))];
  hbuf[t] = v > 0.f ? v : expm1f(v);
}

// ------------------------------ edge passes, layer 2 ------------------------------
__global__ void edge2_pass1(const int* __restrict__ ei, int E, int Etot,
                            const float* __restrict__ as2, const float* __restrict__ ad2,
                            float* __restrict__ denom2) {
  int idx = blockIdx.x * blockDim.x + threadIdx.x;
  if (idx >= Etot) return;
  int s, d; edge_nodes(ei, E, idx, s, d);
  float e = as2[s] + ad2[d];
  e = e > 0.f ? e : SLOPE * e;
  atomicAdd(&denom2[d], expf(e));
}

__global__ void edge2_pass2(const int* __restrict__ ei, int E, int Etot,
                            const float* __restrict__ as2, const float* __restrict__ ad2,
                            const float* __restrict__ denom2, const float* __restrict__ h2,
                            float* __restrict__ out2) {
  int t = blockIdx.x * blockDim.x + threadIdx.x;   // (edge, quarter-row)
  int idx = t >> 2, q = t & 3;
  if (idx >= Etot) return;
  int s, d; edge_nodes(ei, E, idx, s, d);
  float e = as2[s] + ad2[d];
  e = e > 0.f ? e : SLOPE * e;
  float alpha = expf(e) / denom2[d];
  const float* __restrict__ hs = h2 + (size_t)s * NCLASS + q * 4;
  float* __restrict__ od = out2 + (size_t)d * NCLASS + q * 4;
  #pragma unroll
  for (int c = 0; c < 4; ++c) atomicAdd(&od[c], hs[c] * alpha);
}

__global__ void final_kernel(const float* __restrict__ out2, const float* __restrict__ side2,
                             const float* __restrict__ b2, const float* __restrict__ fc2b,
                             float* __restrict__ out, int N) {
  int n = blockIdx.x * blockDim.x + threadIdx.x;
  if (n >= N) return;
  float z[NCLASS], m = -INFINITY;
  #pragma unroll
  for (int c = 0; c < NCLASS; ++c) {
    z[c] = out2[(size_t)n * NCLASS + c] + b2[c]
         - LAMBDA2 * (side2[(size_t)n * NCLASS + c] + fc2b[c]);
    m = fmaxf(m, z[c]);
  }
  float s = 0.f;
  #pragma unroll
  for (int c = 0; c < NCLASS; ++c) s += expf(z[c] - m);
  float lse = m + logf(s);
  #pragma unroll
  for (int c = 0; c < NCLASS; ++c) out[(size_t)n * NCLASS + c] = z[c] - lse;
}

extern "C" void kernel_launch(void* const* d_in, const int* in_sizes, int n_in,
                              void* d_out, int out_size, void* d_ws, size_t ws_size,
                              hipStream_t stream) {
  const float* x    = (const float*)d_in[0];
  const int*   ei   = (const int*)  d_in[1];
  const float* W1   = (const float*)d_in[2];
  const float* a1s  = (const float*)d_in[3];
  const float* a1d  = (const float*)d_in[4];
  const float* b1   = (const float*)d_in[5];
  const float* W2   = (const float*)d_in[6];
  const float* a2s  = (const float*)d_in[7];
  const float* a2d  = (const float*)d_in[8];
  const float* b2   = (const float*)d_in[9];
  const float* fc1W = (const float*)d_in[10];
  const float* fc1b = (const float*)d_in[11];
  const float* fc2W = (const float*)d_in[12];
  const float* fc2b = (const float*)d_in[13];

  const int N    = in_sizes[0] / NFEAT;
  const int E    = in_sizes[1] / 2;
  const int Etot = E + N;

  char* p = (char*)d_ws;
  auto carve = [&](size_t nfloats) {
    float* r = (float*)p;
    p += ((nfloats * sizeof(float) + 255) / 256) * 256;
    return r;
  };
  float* h1     = carve((size_t)N * HD);
  float* as1    = carve((size_t)N * HEADS);
  float* ad1    = carve((size_t)N * HEADS);
  float* side1  = carve((size_t)N * NHID);
  float* denom1 = carve((size_t)N * HEADS);
  float* out1   = carve((size_t)N * HD);
  float* hbuf   = carve((size_t)N * HD);
  float* h2     = carve((size_t)N * NCLASS);
  float* side2  = carve((size_t)N * NCLASS);
  float* as2    = carve((size_t)N);
  float* ad2    = carve((size_t)N);
  float* denom2 = carve((size_t)N);
  float* out2   = carve((size_t)N * NCLASS);

  hipMemsetAsync(denom1, 0, (size_t)N * HEADS * sizeof(float), stream);
  hipMemsetAsync(out1,   0, (size_t)N * HD    * sizeof(float), stream);
  hipMemsetAsync(denom2, 0, (size_t)N         * sizeof(float), stream);
  hipMemsetAsync(out2,   0, (size_t)N * NCLASS* sizeof(float), stream);

  const int mtiles = N / 16;                // N = 50000 = 16 * 3125

  gemm1_wmma<<<(mtiles + 1) / 2, 256, 0, stream>>>(x, W1, h1, N, mtiles);
  attn1_kernel<<<(N + 127) / 128, 128, 0, stream>>>(x, h1, fc1W, fc1b, a1s, a1d,
                                                    as1, ad1, side1, N);
  edge1_pass1<<<(Etot + 255) / 256, 256, 0, stream>>>(ei, E, Etot, as1, ad1, denom1);
  {
    long long t = (long long)Etot * HEADS;
    edge1_pass2<<<(unsigned)((t + 255) / 256), 256, 0, stream>>>(ei, E, Etot, as1, ad1,
                                                                 denom1, h1, out1);
  }
  {
    long long t = (long long)N * HD;
    elu_kernel<<<(unsigned)((t + 255) / 256), 256, 0, stream>>>(out1, side1, b1, hbuf, t);
  }
  gemm2_wmma<<<(mtiles + 7) / 8, 256, 0, stream>>>(hbuf, W2, fc2W, h2, side2, N, mtiles);
  attn2_kernel<<<(N + 255) / 256, 256, 0, stream>>>(h2, a2s, a2d, as2, ad2, N);
  edge2_pass1<<<(Etot + 255) / 256, 256, 0, stream>>>(ei, E, Etot, as2, ad2, denom2);
  {
    long long t = (long long)Etot * 4;
    edge2_pass2<<<(unsigned)((t + 255) / 256), 256, 0, stream>>>(ei, E, Etot, as2, ad2,
                                                                 denom2, h2, out2);
  }
  final_kernel<<<(N + 255) / 256, 256, 0, stream>>>(out2, side2, b2, fc2b,
                                                    (float*)d_out, N);
}